// PauliOperator_41669772706628
// MI455X (gfx1250) — compile-verified
//
#include <hip/hip_runtime.h>

// Pauli-Y on qubit 5 of a 22-qubit state, batch 8.
// x viewed as [8, 32, 2, 65536] (floats per re/im plane):
//   y0 = -i*x1  -> (re,im) = ( im1, -re1)
//   y1 =  i*x0  -> (re,im) = (-im0,  re0)
// Output is complex64 interleaved: out_float[2k] = re, out_float[2k+1] = im.
// Pure streaming permute: 512 MiB of traffic, HBM-bound (~22 us at 23.3 TB/s).

typedef float __attribute__((ext_vector_type(4))) v4f;
typedef int   __attribute__((ext_vector_type(4))) v4i;

namespace {
constexpr unsigned kRight4  = 16384;            // 2^16 floats / 4 per slab half
constexpr unsigned kSlab4   = 2 * kRight4;      // vec4 per (b,a) slab
constexpr unsigned kTiles   = 16384;            // total vec4-pairs / 256
constexpr unsigned kThreads = 256;              // 8 waves (wave32)
constexpr unsigned kBlocks  = 4096;             // 4 tiles per block -> pipelining
}

#if defined(__HIP_DEVICE_COMPILE__) && defined(__gfx1250__) &&                 \
    __has_builtin(__builtin_amdgcn_global_load_async_to_lds_b128) &&           \
    __has_builtin(__builtin_amdgcn_s_wait_asynccnt)
#define PAULI_ASYNC 1
#else
#define PAULI_ASYNC 0
#endif

__device__ __forceinline__ void pauli_emit(v4f* __restrict__ out,
                                           unsigned i0, unsigned i1,
                                           v4f r0, v4f m0, v4f r1, v4f m1) {
  // y0 at complex indices [4*i0 .. 4*i0+3]
  v4f o;
  o = v4f{m1.x, -r1.x, m1.y, -r1.y};
  __builtin_nontemporal_store(o, out + 2u * i0);
  o = v4f{m1.z, -r1.z, m1.w, -r1.w};
  __builtin_nontemporal_store(o, out + 2u * i0 + 1u);
  // y1 at complex indices [4*i1 .. 4*i1+3]
  o = v4f{-m0.x, r0.x, -m0.y, r0.y};
  __builtin_nontemporal_store(o, out + 2u * i1);
  o = v4f{-m0.z, r0.z, -m0.w, r0.w};
  __builtin_nontemporal_store(o, out + 2u * i1 + 1u);
}

__device__ __forceinline__ void pair_index(unsigned tile, unsigned t,
                                           unsigned& i0, unsigned& i1) {
  unsigned p    = tile * kThreads + t;      // vec4-pair index
  unsigned slab = p >> 14;                  // / kRight4
  unsigned c    = p & (kRight4 - 1u);
  i0 = slab * kSlab4 + c;                   // qubit bit = 0
  i1 = i0 + kRight4;                        // qubit bit = 1
}

__global__ __launch_bounds__(kThreads) void pauliy_kernel(
    const v4f* __restrict__ re, const v4f* __restrict__ im,
    v4f* __restrict__ out) {
  const unsigned t = threadIdx.x;

#if PAULI_ASYNC
  // Depth-2 async->LDS pipeline. Each lane stages its own 4x16B, so no
  // barrier is required: only s_wait_asynccnt orders producer/consumer.
  __shared__ v4f stage[2][4][kThreads];     // 32 KiB of 320 KiB

  auto issue = [&](unsigned tile, unsigned buf) {
    unsigned i0, i1;
    pair_index(tile, t, i0, i1);
    __builtin_amdgcn_global_load_async_to_lds_b128(
        (__attribute__((address_space(1))) v4i*)(re + i0),
        (__attribute__((address_space(3))) v4i*)&stage[buf][0][t], 0, 0);
    __builtin_amdgcn_global_load_async_to_lds_b128(
        (__attribute__((address_space(1))) v4i*)(im + i0),
        (__attribute__((address_space(3))) v4i*)&stage[buf][1][t], 0, 0);
    __builtin_amdgcn_global_load_async_to_lds_b128(
        (__attribute__((address_space(1))) v4i*)(re + i1),
        (__attribute__((address_space(3))) v4i*)&stage[buf][2][t], 0, 0);
    __builtin_amdgcn_global_load_async_to_lds_b128(
        (__attribute__((address_space(1))) v4i*)(im + i1),
        (__attribute__((address_space(3))) v4i*)&stage[buf][3][t], 0, 0);
  };

  unsigned tile = blockIdx.x;               // < kBlocks <= kTiles
  unsigned buf  = 0;
  issue(tile, buf);
  for (;;) {
    unsigned next     = tile + kBlocks;
    bool     has_next = next < kTiles;
    if (has_next) {
      issue(next, buf ^ 1u);
      __builtin_amdgcn_s_wait_asynccnt(4);  // 4 in flight = next tile only
    } else {
      __builtin_amdgcn_s_wait_asynccnt(0);
    }
    asm volatile("" ::: "memory");          // keep DS reads after the wait

    unsigned i0, i1;
    pair_index(tile, t, i0, i1);
    v4f r0 = stage[buf][0][t];
    v4f m0 = stage[buf][1][t];
    v4f r1 = stage[buf][2][t];
    v4f m1 = stage[buf][3][t];
    pauli_emit(out, i0, i1, r0, m0, r1, m1);

    if (!has_next) break;
    tile = next;
    buf ^= 1u;
  }
#else
  // Fallback: direct nontemporal b128 streaming (still bandwidth-optimal).
  for (unsigned tile = blockIdx.x; tile < kTiles; tile += kBlocks) {
    unsigned i0, i1;
    pair_index(tile, t, i0, i1);
    __builtin_prefetch(re + i0 + kBlocks * kThreads, 0, 0);
    v4f r0 = __builtin_nontemporal_load(re + i0);
    v4f m0 = __builtin_nontemporal_load(im + i0);
    v4f r1 = __builtin_nontemporal_load(re + i1);
    v4f m1 = __builtin_nontemporal_load(im + i1);
    pauli_emit(out, i0, i1, r0, m0, r1, m1);
  }
#endif
}

extern "C" void kernel_launch(void* const* d_in, const int* in_sizes, int n_in,
                              void* d_out, int out_size, void* d_ws,
                              size_t ws_size, hipStream_t stream) {
  (void)in_sizes; (void)n_in; (void)out_size; (void)d_ws; (void)ws_size;
  const v4f* re = (const v4f*)d_in[0];
  const v4f* im = (const v4f*)d_in[1];
  v4f* out = (v4f*)d_out;
  pauliy_kernel<<<kBlocks, kThreads, 0, stream>>>(re, im, out);
}